// SpatialEncoder_18330920419432
// MI455X (gfx1250) — compile-verified
//
#include <hip/hip_runtime.h>

// ---------------- problem constants ----------------
#define B_   16
#define E_   512
#define D_   256
#define W_   128
#define MC_  18

typedef __attribute__((ext_vector_type(2))) float v2f;
typedef __attribute__((ext_vector_type(8))) float v8f;

// One fp32 WMMA step: D(16x16) += A(16x4) * B(4x16), full fp32.
// A layout (ISA 7.12.2): lane l supplies A[m=l&15][kbase + (l>>4)*2 + j], j=0,1
// B layout (row-striped): lane l supplies B[kbase + (l>>4)*2 + j][n=l&15]
__device__ __forceinline__ v8f wmma4(float a0, float a1, float b0, float b1, v8f c) {
  v2f a; a[0] = a0; a[1] = a1;
  v2f b; b[0] = b0; b[1] = b1;
  return __builtin_amdgcn_wmma_f32_16x16x4_f32(false, a, false, b, (short)0, c,
                                               false, false);
}

__global__ void zero_kernel(float* __restrict__ p, int n) {
  int i = blockIdx.x * blockDim.x + threadIdx.x;
  if (i < n) p[i] = 0.0f;
}

__global__ void copy_kernel(const float* __restrict__ a, float* __restrict__ b, int n) {
  int i = blockIdx.x * blockDim.x + threadIdx.x;
  if (i < n) b[i] = a[i];
}

// ---------------- scatter-add: one wave per entity, one lane per channel ----------------
__global__ void scatter_kernel(const float* __restrict__ exy,
                               const float* __restrict__ red,
                               float* __restrict__ smap) {
  int t = blockIdx.x * blockDim.x + threadIdx.x;
  int ent = t >> 5;           // 0..8191
  int c = t & 31;
  int b = ent >> 9;           // / E_
  const float* bits = exy + (size_t)ent * 16;
  float xv = 0.0f, yv = 0.0f;
  #pragma unroll
  for (int i = 0; i < 8; ++i) { xv = xv * 2.0f + bits[i]; yv = yv * 2.0f + bits[8 + i]; }
  float valid = (bits[0] != -1e9f) ? 1.0f : 0.0f;
  int xi = (int)(xv * 0.25f); xi = xi < 0 ? 0 : (xi > W_ - 1 ? W_ - 1 : xi);
  int yi = (int)(yv * 0.25f); yi = yi < 0 ? 0 : (yi > W_ - 1 ? W_ - 1 : yi);
  float val = red[(size_t)ent * 32 + c] * valid;
  atomicAdd(&smap[(((size_t)b * W_ + yi) * W_ + xi) * 32 + c], val);
}

// ---------------- project 1x1 conv over concat(smap[32], x[18]); 2x2 tile blocking ----------------
__global__ void project_kernel(const float* __restrict__ smap,
                               const float* __restrict__ xin,
                               const float* __restrict__ wgt,
                               const float* __restrict__ bias,
                               float* __restrict__ out) {
  const int N = 32;
  int wave = blockIdx.x * (blockDim.x >> 5) + (threadIdx.x >> 5);  // m-group id
  int lane = threadIdx.x & 31;
  int m0 = wave * 32;
  int mA = lane & 15, kg = (lane >> 4) * 2, nB = lane & 15;
  int bA[2], yA[2], xA[2];
  const float* sb[2];
  #pragma unroll
  for (int mi = 0; mi < 2; ++mi) {
    int m = m0 + mi * 16 + mA;
    bA[mi] = m >> 14; int pix = m & 16383; yA[mi] = pix >> 7; xA[mi] = pix & 127;
    sb[mi] = smap + ((((size_t)bA[mi] * W_) + yA[mi]) * W_ + xA[mi]) * 32;
  }
  v8f acc[2][2] = {};
  for (int k = 0; k < 52; k += 4) {
    int k0 = k + kg, k1 = k0 + 1;
    float a0[2], a1[2];
    #pragma unroll
    for (int mi = 0; mi < 2; ++mi) {
      a0[mi] = (k0 < 32) ? sb[mi][k0]
             : ((k0 < 50) ? xin[(((size_t)bA[mi] * MC_ + (k0 - 32)) * W_ + yA[mi]) * W_ + xA[mi]] : 0.0f);
      a1[mi] = (k1 < 32) ? sb[mi][k1]
             : ((k1 < 50) ? xin[(((size_t)bA[mi] * MC_ + (k1 - 32)) * W_ + yA[mi]) * W_ + xA[mi]] : 0.0f);
    }
    #pragma unroll
    for (int ni = 0; ni < 2; ++ni) {
      float b0 = (k0 < 50) ? wgt[k0 * N + ni * 16 + nB] : 0.0f;
      float b1 = (k1 < 50) ? wgt[k1 * N + ni * 16 + nB] : 0.0f;
      #pragma unroll
      for (int mi = 0; mi < 2; ++mi)
        acc[mi][ni] = wmma4(a0[mi], a1[mi], b0, b1, acc[mi][ni]);
    }
  }
  int mrow = 8 * (lane >> 4);
  #pragma unroll
  for (int mi = 0; mi < 2; ++mi)
    #pragma unroll
    for (int ni = 0; ni < 2; ++ni)
      #pragma unroll
      for (int r = 0; r < 8; ++r) {
        float v = acc[mi][ni][r] + bias[ni * 16 + nB];
        v = v > 0.0f ? v : 0.0f;
        out[(size_t)(m0 + mi * 16 + mrow + r) * N + ni * 16 + nB] = v;
      }
}

// ---------------- generic implicit-GEMM conv (NHWC in, HWIO weights), MBLKxNBLK blocking ----
// EPI: 0 = bias + relu
//      1 = bn + relu
//      2 = bn + residual(in-place on out) + relu, skip += result
//      3 = relu only (plain GEMM)
template <int KH, int KW, int S, int PAD, int CIN, int COUT,
          int HIN, int WIN, int HOUT, int WOUT, int EPI, int MBLK, int NBLK>
__global__ void conv_wmma_kernel(const float* __restrict__ in,
                                 const float* __restrict__ wgt,
                                 const float* __restrict__ bias,
                                 const float* __restrict__ bng,
                                 const float* __restrict__ bnb,
                                 const float* __restrict__ bnm,
                                 const float* __restrict__ bnv,
                                 float* __restrict__ skip,
                                 float* __restrict__ out) {
  const int K = KH * KW * CIN;
  const int N = COUT;
  const int tilesNg = N / (16 * NBLK);
  int wave = blockIdx.x * (blockDim.x >> 5) + (threadIdx.x >> 5);
  int lane = threadIdx.x & 31;
  int mt = wave / tilesNg, nt = wave % tilesNg;
  int m0 = mt * 16 * MBLK, n0 = nt * 16 * NBLK;
  int mA = lane & 15, kg = (lane >> 4) * 2, nB = lane & 15;
  int bA[MBLK], oyA[MBLK], oxA[MBLK];
  #pragma unroll
  for (int mi = 0; mi < MBLK; ++mi) {
    int m = m0 + mi * 16 + mA;
    bA[mi] = m / (HOUT * WOUT);
    int rem = m % (HOUT * WOUT);
    oyA[mi] = rem / WOUT; oxA[mi] = rem % WOUT;
  }
  v8f acc[MBLK][NBLK] = {};
  for (int k = 0; k < K; k += 4) {
    int k0 = k + kg, k1 = k0 + 1;
    int ky0 = k0 / (KW * CIN); int r20 = k0 % (KW * CIN);
    int kx0 = r20 / CIN;       int ci0 = r20 % CIN;
    int ky1 = k1 / (KW * CIN); int r21 = k1 % (KW * CIN);
    int kx1 = r21 / CIN;       int ci1 = r21 % CIN;
    float a0[MBLK], a1[MBLK];
    #pragma unroll
    for (int mi = 0; mi < MBLK; ++mi) {
      int iy0 = oyA[mi] * S + ky0 - PAD, ix0 = oxA[mi] * S + kx0 - PAD;
      a0[mi] = (iy0 >= 0 && iy0 < HIN && ix0 >= 0 && ix0 < WIN)
                   ? in[(((size_t)bA[mi] * HIN + iy0) * WIN + ix0) * CIN + ci0] : 0.0f;
      int iy1 = oyA[mi] * S + ky1 - PAD, ix1 = oxA[mi] * S + kx1 - PAD;
      a1[mi] = (iy1 >= 0 && iy1 < HIN && ix1 >= 0 && ix1 < WIN)
                   ? in[(((size_t)bA[mi] * HIN + iy1) * WIN + ix1) * CIN + ci1] : 0.0f;
    }
    #pragma unroll
    for (int ni = 0; ni < NBLK; ++ni) {
      float b0 = wgt[(size_t)k0 * N + n0 + ni * 16 + nB];
      float b1 = wgt[(size_t)k1 * N + n0 + ni * 16 + nB];
      #pragma unroll
      for (int mi = 0; mi < MBLK; ++mi)
        acc[mi][ni] = wmma4(a0[mi], a1[mi], b0, b1, acc[mi][ni]);
    }
  }
  int mrow = 8 * (lane >> 4);
  #pragma unroll
  for (int mi = 0; mi < MBLK; ++mi)
    #pragma unroll
    for (int ni = 0; ni < NBLK; ++ni)
      #pragma unroll
      for (int r = 0; r < 8; ++r) {
        int mm2 = m0 + mi * 16 + mrow + r;
        int cidx = n0 + ni * 16 + nB;
        size_t idx = (size_t)mm2 * N + cidx;
        float v = acc[mi][ni][r];
        if constexpr (EPI == 0) {
          v += bias[cidx];
          v = v > 0.0f ? v : 0.0f;
          out[idx] = v;
        } else if constexpr (EPI == 3) {
          v = v > 0.0f ? v : 0.0f;
          out[idx] = v;
        } else {
          v = (v - bnm[cidx]) * rsqrtf(bnv[cidx] + 1e-3f) * bng[cidx] + bnb[cidx];
          if constexpr (EPI == 2) {
            v += out[idx];            // residual, in-place safe (1 lane owns idx)
            v = v > 0.0f ? v : 0.0f;
            out[idx] = v;
            skip[idx] += v;
          } else {
            v = v > 0.0f ? v : 0.0f;
            out[idx] = v;
          }
        }
      }
}

// ---------------- skip NHWC [16,16,16,128] -> NCHW [16,128,16,16] ----------------
__global__ void transpose_skip_kernel(const float* __restrict__ skip,
                                      float* __restrict__ out) {
  int i = blockIdx.x * blockDim.x + threadIdx.x;   // 0..524287
  int b = i >> 15;
  int rem = i & 32767;
  int c = rem >> 8;
  int p = rem & 255;
  int y = p >> 4, x = p & 15;
  out[i] = skip[((((size_t)b * 16) + y) * 16 + x) * 128 + c];
}

// ---------------- FC: relu(flatNCHW(h)[16,32768] @ fc_w[32768,256] + b) ----------------
__global__ void fc_kernel(const float* __restrict__ h,
                          const float* __restrict__ wgt,
                          const float* __restrict__ bias,
                          float* __restrict__ out) {
  const int N = 256, K = 32768;
  int wave = blockIdx.x * (blockDim.x >> 5) + (threadIdx.x >> 5);  // 0..15
  int lane = threadIdx.x & 31;
  int n0 = wave * 16;
  int mA = lane & 15, kg = (lane >> 4) * 2, nB = lane & 15;
  v8f acc = {};
  for (int k = 0; k < K; k += 4) {
    int k0 = k + kg, k1 = k0 + 1;
    int c0 = k0 >> 8, y0 = (k0 >> 4) & 15, x0 = k0 & 15;
    int c1 = k1 >> 8, y1 = (k1 >> 4) & 15, x1 = k1 & 15;
    float a0 = h[((((size_t)mA * 16) + y0) * 16 + x0) * 128 + c0];
    float a1 = h[((((size_t)mA * 16) + y1) * 16 + x1) * 128 + c1];
    float b0 = wgt[(size_t)k0 * N + n0 + nB];
    float b1 = wgt[(size_t)k1 * N + n0 + nB];
    acc = wmma4(a0, a1, b0, b1, acc);
  }
  int mrow = 8 * (lane >> 4);
  #pragma unroll
  for (int r = 0; r < 8; ++r) {
    float v = acc[r] + bias[n0 + nB];
    v = v > 0.0f ? v : 0.0f;
    out[(size_t)(mrow + r) * N + n0 + nB] = v;
  }
}

extern "C" void kernel_launch(void* const* d_in, const int* in_sizes, int n_in,
                              void* d_out, int out_size, void* d_ws, size_t ws_size,
                              hipStream_t stream) {
  const float* xin = (const float*)d_in[0];
  const float* emb = (const float*)d_in[1];
  const float* exy = (const float*)d_in[2];
  const float* c1w = (const float*)d_in[3];
  const float* pw  = (const float*)d_in[4];
  const float* pb  = (const float*)d_in[5];
  const float* d1w = (const float*)d_in[6];
  const float* d1b = (const float*)d_in[7];
  const float* d2w = (const float*)d_in[8];
  const float* d2b = (const float*)d_in[9];
  const float* d3w = (const float*)d_in[10];
  const float* d3b = (const float*)d_in[11];
  const float* rw1 = (const float*)d_in[12];
  const float* rw2 = (const float*)d_in[13];
  const float* b1g = (const float*)d_in[14];
  const float* b1b = (const float*)d_in[15];
  const float* b1m = (const float*)d_in[16];
  const float* b1v = (const float*)d_in[17];
  const float* b2g = (const float*)d_in[18];
  const float* b2b = (const float*)d_in[19];
  const float* b2m = (const float*)d_in[20];
  const float* b2v = (const float*)d_in[21];
  const float* fcw = (const float*)d_in[22];
  const float* fcb = (const float*)d_in[23];
  float* out = (float*)d_out;

  // workspace layout (floats), with region reuse: total ~74.4 MB
  float* ws = (float*)d_ws;
  float* reduced = ws;                       // 8192*32       =   262144
  float* regA = reduced + 262144;            // smap, then h2 =  8388608
  float* regB = regA + 8388608;              // h1, then h3   =  8388608
  float* h4   = regB + 8388608;              // 16*16*16*128  =   524288
  float* obuf = h4 + 524288;                 //                   524288
  float* skip = obuf + 524288;               //                   524288

  // 1) zero the scatter map
  zero_kernel<<<8388608 / 256, 256, 0, stream>>>(regA, 8388608);
  // 2) entity projection GEMM (WMMA, 2x2 blocked) + relu: [8192,256]@[256,32]
  //    expressed as 1x1 conv with B*H*W = 8192 (b index == entity batch)
  conv_wmma_kernel<1,1,1,0, 256, 32, 32, 16, 32, 16, 3, 2, 2>
      <<<64, 128, 0, stream>>>(emb, c1w, nullptr, nullptr, nullptr, nullptr, nullptr,
                               nullptr, reduced);
  // 3) scatter-add entities onto the map
  scatter_kernel<<<262144 / 256, 256, 0, stream>>>(exy, reduced, regA);
  // 4) project 1x1 conv over concat(smap, x) -> h1 (regB)
  project_kernel<<<2048, 128, 0, stream>>>(regA, xin, pw, pb, regB);
  // 5) downsample convs (WMMA implicit GEMM, 2x2 blocked), SAME pad = 1 for 4x4/s2
  conv_wmma_kernel<4,4,2,1, 32, 64,128,128, 64, 64, 0, 2, 2>
      <<<1024, 128, 0, stream>>>(regB, d1w, d1b, nullptr, nullptr, nullptr, nullptr,
                                 nullptr, regA);
  conv_wmma_kernel<4,4,2,1, 64,128, 64, 64, 32, 32, 0, 2, 2>
      <<<512, 128, 0, stream>>>(regA, d2w, d2b, nullptr, nullptr, nullptr, nullptr,
                                nullptr, regB);
  conv_wmma_kernel<4,4,2,1,128,128, 32, 32, 16, 16, 0, 2, 2>
      <<<128, 128, 0, stream>>>(regB, d3w, d3b, nullptr, nullptr, nullptr, nullptr,
                                nullptr, h4);
  // 6) skip = h
  copy_kernel<<<524288 / 256, 256, 0, stream>>>(h4, skip, 524288);
  // 7) residual blocks (3x3, SAME pad = 1)
  for (int i = 0; i < 4; ++i) {
    const float* w1 = rw1 + (size_t)i * 3 * 3 * 128 * 128;
    const float* w2 = rw2 + (size_t)i * 3 * 3 * 128 * 128;
    conv_wmma_kernel<3,3,1,1,128,128,16,16,16,16,1,2,2>
        <<<128, 128, 0, stream>>>(h4, w1, nullptr,
                                  b1g + i * 128, b1b + i * 128,
                                  b1m + i * 128, b1v + i * 128,
                                  nullptr, obuf);
    conv_wmma_kernel<3,3,1,1,128,128,16,16,16,16,2,2,2>
        <<<128, 128, 0, stream>>>(obuf, w2, nullptr,
                                  b2g + i * 128, b2b + i * 128,
                                  b2m + i * 128, b2v + i * 128,
                                  skip, h4);
  }
  // 8) map_skip output (NCHW)
  transpose_skip_kernel<<<2048, 256, 0, stream>>>(skip, out);
  // 9) embedded_spatial = relu(fc)
  fc_kernel<<<4, 128, 0, stream>>>(h4, fcw, fcb, out + 524288);
}